// EncoderPositional_31129922961550
// MI455X (gfx1250) — compile-verified
//
#include <hip/hip_runtime.h>

typedef float v2f __attribute__((ext_vector_type(2)));
typedef float v4f __attribute__((ext_vector_type(4)));
typedef float v8f __attribute__((ext_vector_type(8)));
typedef unsigned int v4u __attribute__((ext_vector_type(4)));
typedef int v4i __attribute__((ext_vector_type(4)));
typedef int v8i __attribute__((ext_vector_type(8)));

#define VOCAB   128000
#define MAXLEN  8192
#define DWORD   512
#define HIDDEN  1024   // DWORD + DPOS

#define KCHUNK    32
#define LDS_STRIDE 36                       // 32 data floats + 4 pad floats per row
#define LDS_BUF_FLOATS (128 * LDS_STRIDE)   // 18432 B per buffer

#if __has_builtin(__builtin_amdgcn_tensor_load_to_lds)
#define USE_TDM 1
#else
#define USE_TDM 0
#endif

#if USE_TDM
// One TDM descriptor load: W[rowBase..rowBase+127][k0..k0+31] -> LDS @ ldsOff,
// hardware-padded to a 36-float row stride (pad 4 DWORDs every 32 DWORDs).
static __device__ __forceinline__ void tdm_load_w_chunk(const float* gsrc, unsigned ldsOff) {
    const unsigned long long ga = (unsigned long long)(uintptr_t)gsrc;
    v4u g0;
    g0.x = 1u;                                        // count=1, user descriptor, no gather
    g0.y = ldsOff;                                    // lds_addr (bytes)
    g0.z = (unsigned)(ga & 0xFFFFFFFFu);              // global_addr[31:0]   (bits 95:64)
    g0.w = (unsigned)((ga >> 32) & 0x01FFFFFFu)       // global_addr[56:32]  (bits 120:96)
         | 0x80000000u;                               // type=2 ("image")    (bits 127:126)
    v8i g1;
    g1[0] = (2 << 16)                                 // data_size = 4 bytes
          | (1 << 20)                                 // pad_enable
          | (4 << 22)                                 // pad_interval: 2^(4+1)=32 DWORDs
          | (3 << 25);                                // pad_amount: 3+1 = 4 DWORDs
    g1[1] = (int)(1024u << 16);                       // tensor_dim0 = 1024 (bits 79:48, low half)
    g1[2] = (int)(1024u << 16);                       // dim0 hi=0 | tensor_dim1 = 1024 (low half)
    g1[3] = (int)(32u << 16);                         // dim1 hi=0 | tile_dim0 = 32
    g1[4] = 128;                                      // tile_dim1 = 128, tile_dim2 = 0
    g1[5] = 1024;                                     // tensor_dim0_stride[31:0] = 1024
    g1[6] = 0;                                        // stride hi | tensor_dim1_stride lo (unused 2D)
    g1[7] = 0;
    const v4i gz4 = {0, 0, 0, 0};                     // groups 2/3: dims 2..4 unused
    const v8i gz8 = {0, 0, 0, 0, 0, 0, 0, 0};        // extra group (6-arg toolchain form)
    __builtin_amdgcn_tensor_load_to_lds(g0, g1, gz4, gz4, gz8, 0);
}
#endif

// 128x128 block tile, 8 waves: 4 along M (32 rows each), 2 along N (64 cols each).
// Each wave: 2 M-subtiles x 4 N-subtiles of 16x16 f32 WMMA accumulators.
__global__ __launch_bounds__(256)
void encpos_gemm(const int* __restrict__ inputs,
                 const float* __restrict__ word_emb,
                 const float* __restrict__ pos_emb,
                 const float* __restrict__ W,
                 const float* __restrict__ bias,
                 float* __restrict__ out) {
    __shared__ __align__(16) float Wlds[2][LDS_BUF_FLOATS];

    const int tid   = threadIdx.x;
    const int lane  = tid & 31;
    const int wave  = tid >> 5;
    const int waveM = wave >> 1;              // 0..3
    const int waveN = wave & 1;               // 0..1
    const int nBlk  = blockIdx.x * 128;
    const int mBlk  = blockIdx.y * 128;
    const int mBase = mBlk + waveM * 32;
    const int l15   = lane & 15;
    const int kLane = (lane >> 4) * 2;        // A/B fragment k offset per lane half
    const int rowHalf = (lane >> 4) * 8;      // C/D fragment row offset per lane half

    // Fused gather: per-lane row base pointers for the two M-subtiles.
    const int m0 = mBase + l15;
    const int m1 = m0 + 16;
    const float* weRow0 = word_emb + (size_t)inputs[m0] * DWORD;
    const float* weRow1 = word_emb + (size_t)inputs[m1] * DWORD;
    const float* peRow0 = pos_emb  + (size_t)m0 * DWORD;
    const float* peRow1 = pos_emb  + (size_t)m1 * DWORD;

    const float* Wtile = W + (size_t)nBlk * HIDDEN;

    v8f acc[2][4] = {};

#if USE_TDM
    const unsigned lds0 = (unsigned)(uintptr_t)&Wlds[0][0];
    const unsigned lds1 = (unsigned)(uintptr_t)&Wlds[1][0];
    // Prime the pipeline: chunk 0 -> buffer 0.
    if (wave == 0) tdm_load_w_chunk(Wtile, lds0);
#endif

    for (int c = 0; c < HIDDEN / KCHUNK; ++c) {
        const int k0 = c * KCHUNK;

#if USE_TDM
        __syncthreads();   // everyone done reading buf[(c+1)&1] (chunk c-1)
        if (wave == 0) {
            if (c + 1 < HIDDEN / KCHUNK) {
                tdm_load_w_chunk(Wtile + (k0 + KCHUNK),
                                 ((c + 1) & 1) ? lds1 : lds0);
                __builtin_amdgcn_s_wait_tensorcnt(1);  // chunk c landed; c+1 may stream
            } else {
                __builtin_amdgcn_s_wait_tensorcnt(0);
            }
        }
        __syncthreads();   // release buf[c&1] to all waves
        const float* Wl = Wlds[c & 1];
#else
        __syncthreads();
        for (int i = tid; i < 128 * 8; i += 256) {
            const int r  = i >> 3;
            const int c4 = (i & 7) << 2;
            const float* src = Wtile + (size_t)r * HIDDEN + k0 + c4;
            *(v4f*)&Wlds[0][r * LDS_STRIDE + c4] = *(const v4f*)src;
            if (k0 + KCHUNK < HIDDEN) __builtin_prefetch(src + KCHUNK, 0, 0);
        }
        __syncthreads();
        const float* Wl = Wlds[0];
#endif

        // Uniform per-chunk source select (chunk never straddles the 512 boundary).
        const float* a0 = (k0 < DWORD) ? (weRow0 + k0 + kLane) : (peRow0 + (k0 - DWORD) + kLane);
        const float* a1 = (k0 < DWORD) ? (weRow1 + k0 + kLane) : (peRow1 + (k0 - DWORD) + kLane);

        #pragma unroll
        for (int kk = 0; kk < KCHUNK; kk += 4) {
            const v2f aA = *(const v2f*)(a0 + kk);   // A frag, M-subtile 0
            const v2f aB = *(const v2f*)(a1 + kk);   // A frag, M-subtile 1
            #pragma unroll
            for (int s = 0; s < 4; ++s) {
                const int nl = waveN * 64 + s * 16 + l15;
                const v2f bf = *(const v2f*)&Wl[nl * LDS_STRIDE + kk + kLane];  // ds_load b64
                acc[0][s] = __builtin_amdgcn_wmma_f32_16x16x4_f32(
                    false, aA, false, bf, (short)0, acc[0][s], false, false);
                acc[1][s] = __builtin_amdgcn_wmma_f32_16x16x4_f32(
                    false, aB, false, bf, (short)0, acc[1][s], false, false);
            }
        }
    }

    // Epilogue: bias + sigmoid + store.
    #pragma unroll
    for (int s = 0; s < 4; ++s) {
        const int col = nBlk + waveN * 64 + s * 16 + l15;
        const float bv = bias[col];
        #pragma unroll
        for (int mi = 0; mi < 2; ++mi) {
            const int rowB = mBase + mi * 16 + rowHalf;
            #pragma unroll
            for (int r = 0; r < 8; ++r) {
                const float x = acc[mi][s][r] + bv;
                const float sg = 1.0f / (1.0f + __expf(-x));
                out[(size_t)(rowB + r) * HIDDEN + col] = sg;
            }
        }
    }
}

// Deterministic two-stage column mean (no float atomics -> replay-stable).
__global__ __launch_bounds__(256)
void encpos_mean_partial(const float* __restrict__ out, float* __restrict__ ws) {
    const int j  = blockIdx.x * 256 + threadIdx.x;   // column 0..1023
    const int r0 = blockIdx.y * 256;                 // row group 0..31
    float s = 0.f;
    for (int i = 0; i < 256; ++i)
        s += out[(size_t)(r0 + i) * HIDDEN + j];     // coalesced across threads
    ws[(size_t)blockIdx.y * HIDDEN + j] = s;
}

__global__ __launch_bounds__(256)
void encpos_mean_final(const float* __restrict__ ws, float* __restrict__ hidden) {
    const int j = blockIdx.x * 256 + threadIdx.x;    // column 0..1023
    float s = 0.f;
    #pragma unroll
    for (int g = 0; g < 32; ++g)
        s += ws[(size_t)g * HIDDEN + j];
    hidden[j] = s * (1.0f / (float)MAXLEN);
}

extern "C" void kernel_launch(void* const* d_in, const int* in_sizes, int n_in,
                              void* d_out, int out_size, void* d_ws, size_t ws_size,
                              hipStream_t stream) {
    const int*   inputs   = (const int*)  d_in[0];
    const float* word_emb = (const float*)d_in[1];
    const float* pos_emb  = (const float*)d_in[2];
    const float* W        = (const float*)d_in[3];
    const float* bias     = (const float*)d_in[4];
    float* out    = (float*)d_out;
    float* hidden = out + (size_t)MAXLEN * HIDDEN;   // tuple output: [output(8192x1024), hidden(1024)]
    float* ws     = (float*)d_ws;                    // 32*1024 floats of scratch

    dim3 gGemm(HIDDEN / 128, MAXLEN / 128);          // (8, 64)
    encpos_gemm<<<gGemm, 256, 0, stream>>>(inputs, word_emb, pos_emb, W, bias, out);

    dim3 gPart(HIDDEN / 256, MAXLEN / 256);          // (4, 32)
    encpos_mean_partial<<<gPart, 256, 0, stream>>>(out, ws);
    encpos_mean_final<<<HIDDEN / 256, 256, 0, stream>>>(ws, hidden);
}